// SDIMLoss_42657615184207
// MI455X (gfx1250) — compile-verified
//
#include <hip/hip_runtime.h>
#include <hip/hip_bf16.h>
#include <stdint.h>

// ---------------------------------------------------------------------------
// SSIM loss, MI455X (gfx1250).
// Bandwidth-bound: 201MB streamed once -> use CDNA5 async global->LDS copies
// (ASYNCcnt) with non-temporal hint; VALU does the (trivial) math.
// ---------------------------------------------------------------------------

#define S_WAIT_ASYNC(NSTR) asm volatile("s_wait_asynccnt " NSTR ::: "memory")

// Per-lane 16B async copy global -> LDS (gfx1250 GLOBAL_LOAD_ASYNC_TO_LDS_B128,
// GV addressing: 64-bit VGPR address, saddr=off). Non-temporal: streamed once.
__device__ __forceinline__ void async_cp16(uint32_t lds_off, const float* gp) {
  asm volatile("global_load_async_to_lds_b128 %0, %1, off th:TH_LOAD_NT"
               :: "v"(lds_off), "v"(gp)
               : "memory");
}

// ---------------------------------------------------------------------------
// Pass 1: one workgroup per (batch, 16-row strip). 256 threads.
// Thread t owns column chunk (t&127)*4 .. +4 and row phase t>>7.
// 24 iterations x 16B per input, staged through a depth-3 async LDS pipeline.
// ---------------------------------------------------------------------------
__global__ __launch_bounds__(256)
void ssim_pass1(const float* __restrict__ X, const float* __restrict__ Y,
                float* __restrict__ blk) {
  const int t     = threadIdx.x;
  const int sr    = blockIdx.x;   // strip row   0..31
  const int b     = blockIdx.y;   // batch       0..31
  const int col4  = t & 127;      // 16B column chunk within 512-wide row
  const int phase = t >> 7;       // row phase 0/1 (rows phase, phase+2, ...)

  __shared__ float4 bx[3][256];   // depth-3 double... triple buffer, x
  __shared__ float4 by[3][256];   // same for y
  __shared__ float  red[5][256];  // cross-thread reduction staging

  // element index of (b, c=0, row=sr*16+phase, col=col4*4)
  const size_t base = (size_t)b * (3u * 512u * 512u)
                    + ((size_t)sr * 16u + (size_t)phase) * 512u
                    + (size_t)col4 * 4u;
  const float* gx = X + base;
  const float* gy = Y + base;

  // LDS byte offsets of this lane's slots (low 32 bits of flat shared addr)
  const uint32_t lx0 = (uint32_t)(size_t)(&bx[0][t]);
  const uint32_t ly0 = (uint32_t)(size_t)(&by[0][t]);

  float sx = 0.f, sy = 0.f, sxx = 0.f, syy = 0.f, sxy = 0.f;

  // step i (0..23): channel c=i>>3, row-group k=i&7 -> row = phase + 2k
  auto issue = [&](int i) {
    const int c = i >> 3, k = i & 7;
    const size_t eoff = (size_t)c * 262144u + (size_t)k * 1024u; // elements
    const uint32_t loff = (uint32_t)(i % 3) * 4096u;             // 256*16 B
    async_cp16(lx0 + loff, gx + eoff);
    async_cp16(ly0 + loff, gy + eoff);
  };
  auto consume = [&](int i) {
    const int s = i % 3;
    const float4 xv = bx[s][t];
    const float4 yv = by[s][t];
    sx += (xv.x + xv.y) + (xv.z + xv.w);
    sy += (yv.x + yv.y) + (yv.z + yv.w);
    sxx = fmaf(xv.x, xv.x, fmaf(xv.y, xv.y, fmaf(xv.z, xv.z, fmaf(xv.w, xv.w, sxx))));
    syy = fmaf(yv.x, yv.x, fmaf(yv.y, yv.y, fmaf(yv.z, yv.z, fmaf(yv.w, yv.w, syy))));
    sxy = fmaf(xv.x, yv.x, fmaf(xv.y, yv.y, fmaf(xv.z, yv.z, fmaf(xv.w, yv.w, sxy))));
  };

  // depth-3 software pipeline; per-wave ASYNCcnt: 2 async ops per step.
  issue(0);
  issue(1);
  for (int i = 0; i < 22; ++i) {
    issue(i + 2);
    S_WAIT_ASYNC("4");   // 3 steps (6 ops) in flight -> oldest step done
    consume(i);
  }
  S_WAIT_ASYNC("2");     // steps 22,23 in flight -> 22 done
  consume(22);
  S_WAIT_ASYNC("0");     // drain
  consume(23);

  red[0][t] = sx; red[1][t] = sy; red[2][t] = sxx; red[3][t] = syy; red[4][t] = sxy;
  __syncthreads();

  // Block bcol gets contributions from threads {p*128 + bcol*4 + j}.
  if (t < 32) {
    float S0 = 0.f, S1 = 0.f, S2 = 0.f, S3 = 0.f, S4 = 0.f;
#pragma unroll
    for (int p = 0; p < 2; ++p) {
#pragma unroll
      for (int j = 0; j < 4; ++j) {
        const int idx = p * 128 + t * 4 + j;
        S0 += red[0][idx]; S1 += red[1][idx]; S2 += red[2][idx];
        S3 += red[3][idx]; S4 += red[4][idx];
      }
    }
    const float inv = 1.0f / 256.0f;           // reference divides by WS*WS only
    const float C1 = 6.5025f, C2 = 58.5225f;   // (0.01*255)^2, (0.03*255)^2
    const float mx = S0 * inv, my = S1 * inv;
    const float vx = S2 * inv - mx * mx;
    const float vy = S3 * inv - my * my;
    const float cv = S4 * inv - mx * my;
    const float num = (2.0f * mx * my + C1) * (2.0f * cv + C2);
    const float den = (mx * mx + my * my + C1) * (vx + vy + C2);
    blk[((size_t)b * 32u + (size_t)sr) * 32u + (size_t)t] = num / den;
  }
}

// ---------------------------------------------------------------------------
// Pass 2: one workgroup per batch; deterministic tree-reduce 1024 block SSIMs.
// ---------------------------------------------------------------------------
__global__ __launch_bounds__(256)
void ssim_pass2(const float* __restrict__ blk, float* __restrict__ out) {
  const int b = blockIdx.x;
  const int t = threadIdx.x;
  const float* p = blk + (size_t)b * 1024u;
  __shared__ float sm[256];
  sm[t] = ((p[t] + p[t + 256]) + (p[t + 512] + p[t + 768]));
  __syncthreads();
  for (int off = 128; off > 0; off >>= 1) {
    if (t < off) sm[t] += sm[t + off];
    __syncthreads();
  }
  if (t == 0) out[b] = (1.0f - sm[0] * (1.0f / 1024.0f)) * 0.5f;
}

// ---------------------------------------------------------------------------
extern "C" void kernel_launch(void* const* d_in, const int* in_sizes, int n_in,
                              void* d_out, int out_size, void* d_ws, size_t ws_size,
                              hipStream_t stream) {
  (void)in_sizes; (void)n_in; (void)out_size; (void)ws_size;
  const float* X = (const float*)d_in[0];
  const float* Y = (const float*)d_in[1];
  float* blk = (float*)d_ws;     // 32*1024 floats = 128 KB scratch
  float* out = (float*)d_out;    // 32 floats

  dim3 g1(32, 32), b1(256);
  hipLaunchKernelGGL(ssim_pass1, g1, b1, 0, stream, X, Y, blk);
  hipLaunchKernelGGL(ssim_pass2, dim3(32), dim3(256), 0, stream, blk, out);
}